// NerfAccSampler_39625368272911
// MI455X (gfx1250) — compile-verified
//
#include <hip/hip_runtime.h>

#define MAX_SAMPS 256

typedef float f4 __attribute__((ext_vector_type(4)));
typedef int   i4 __attribute__((ext_vector_type(4)));

// ---------------- kernel 0: zero the per-ray segment tables -------------
__global__ void k_init_seg(int* __restrict__ seg_start, int* __restrict__ seg_end,
                           int n_rays) {
  int r = blockIdx.x * blockDim.x + threadIdx.x;
  if (r < n_rays) { seg_start[r] = 0; seg_end[r] = 0; }
}

// ------- kernel 1: segment boundaries (ray_indices is sorted) -----------
// 4 elements per thread via aligned int4 + one scalar peek; each boundary
// is written by exactly one thread -> race-free, no atomics.
__global__ void k_boundaries(const int* __restrict__ ri,
                             int* __restrict__ seg_start,
                             int* __restrict__ seg_end, int T) {
  int g  = blockIdx.x * blockDim.x + threadIdx.x;
  int i0 = g << 2;
  if (i0 >= T) return;
  if (i0 + 4 <= T) {
    i4 a = *(const i4*)(ri + i0);                    // 16B-aligned (i0 % 4 == 0)
    int nxt = (i0 + 4 < T) ? ri[i0 + 4] : -1;
    int cur[5] = {a.x, a.y, a.z, a.w, nxt};
    if (i0 == 0) seg_start[a.x] = 0;
#pragma unroll
    for (int k = 0; k < 4; ++k) {
      int r = cur[k], rn = cur[k + 1];
      if (rn != r) {
        seg_end[r] = i0 + k + 1;                     // exclusive end
        if (i0 + k + 1 < T) seg_start[rn] = i0 + k + 1;
      }
    }
  } else {                                           // tail (generic T)
    for (int k = 0; k < 4 && i0 + k < T; ++k) {
      int i = i0 + k;
      int r = ri[i];
      if (i == 0) seg_start[r] = 0;
      int rn = (i + 1 < T) ? ri[i + 1] : -1;
      if (rn != r) { seg_end[r] = i + 1; if (i + 1 < T) seg_start[rn] = i + 1; }
    }
  }
}

// ------- kernel 2: dense (ray, slot) grids: valid / z_vals / dists ------
// 4 rays per 256-thread block; each thread owns 4 consecutive slots and
// emits one non-temporal b128 store per output array (outputs are write-
// once streams -> NT keeps rays_chunk/t_* resident in cache instead).
__global__ void __launch_bounds__(256)
k_fill(const float* __restrict__ t_starts, const float* __restrict__ t_ends,
       const int* __restrict__ seg_start, const int* __restrict__ seg_end,
       f4* __restrict__ valid4, f4* __restrict__ z4, f4* __restrict__ d4,
       int n_rays) {
  int tid = threadIdx.x;
  int r   = blockIdx.x * 4 + (tid >> 6);
  if (r >= n_rays) return;
  int s0  = (tid & 63) << 2;                         // first of this thread's 4 slots
  int st  = seg_start[r];
  int num = seg_end[r] - st;
  if (num > MAX_SAMPS) num = MAX_SAMPS;              // mode='drop' semantics
  f4 vv, vz, vd;
#pragma unroll
  for (int k = 0; k < 4; ++k) {
    int s = s0 + k;
    float z = 0.0f, d = 0.0f, v = 0.0f;
    if (s < num) {
      float a = t_starts[st + s];
      float b = t_ends[st + s];
      z = 0.5f * (a + b);
      d = b - a;
      v = 1.0f;
    }
    vv[k] = v; vz[k] = z; vd[k] = d;
  }
  size_t o = ((size_t)r * MAX_SAMPS + s0) >> 2;
  __builtin_nontemporal_store(vv, valid4 + o);
  __builtin_nontemporal_store(vz, z4 + o);
  __builtin_nontemporal_store(vd, d4 + o);
}

// ------- kernel 3: per-sample xyz via async global->LDS b128 staging ----
// Threads 0..191 each async-copy one 16B chunk of the block's ri/ts/te
// tile into LDS (6 per-wave GLOBAL_LOAD_ASYNC_TO_LDS_B128 ops total),
// wave waits on ASYNCcnt, barrier, then each thread gathers its ray row
// (L2-resident: sorted indices) and emits one NT b128 store.
__global__ void __launch_bounds__(256)
k_xyz(const float* __restrict__ rays, const int* __restrict__ ri,
      const float* __restrict__ ts, const float* __restrict__ te,
      f4* __restrict__ xyzs, int T) {
  __shared__ int   s_ri[256];
  __shared__ float s_ts[256];
  __shared__ float s_te[256];
  int tid  = threadIdx.x;
  int base = blockIdx.x * 256;
  int i    = base + tid;
  bool full = (base + 256 <= T);                     // uniform per block
  if (full) {
    if (tid < 192) {
      int grp = tid >> 6;                            // 0: ri, 1: ts, 2: te
      int l   = (tid & 63) << 2;                     // element offset in tile
      unsigned lofs;                                 // low 32 bits of flat shared
      const void* g;                                 // address == LDS byte offset
      if (grp == 0)      { lofs = (unsigned)(size_t)&s_ri[l]; g = (const void*)(ri + base + l); }
      else if (grp == 1) { lofs = (unsigned)(size_t)&s_ts[l]; g = (const void*)(ts + base + l); }
      else               { lofs = (unsigned)(size_t)&s_te[l]; g = (const void*)(te + base + l); }
      asm volatile("global_load_async_to_lds_b128 %0, %1, off"
                   :: "v"(lofs), "v"(g) : "memory");
    }
    asm volatile("s_wait_asynccnt 0" ::: "memory");
  } else if (i < T) {                                // tail fallback
    s_ri[tid] = ri[i];
    s_ts[tid] = ts[i];
    s_te[tid] = te[i];
  }
  __syncthreads();
  if (i >= T) return;

  int r = s_ri[tid];
  const float2* rays2 = (const float2*)rays;         // row r = 3 aligned float2
  float2 a = rays2[(size_t)r * 3 + 0];               // (ox, oy)
  float2 b = rays2[(size_t)r * 3 + 1];               // (oz, dx)
  float2 c = rays2[(size_t)r * 3 + 2];               // (dy, dz)
  float pz = 0.5f * (s_ts[tid] + s_te[tid]);
  f4 o;
  o.x = fmaf(pz, b.y, a.x);
  o.y = fmaf(pz, c.x, a.y);
  o.z = fmaf(pz, c.y, b.x);
  o.w = 0.0f;
  __builtin_nontemporal_store(o, xyzs + i);
}

extern "C" void kernel_launch(void* const* d_in, const int* in_sizes, int n_in,
                              void* d_out, int out_size, void* d_ws, size_t ws_size,
                              hipStream_t stream) {
  const float* rays     = (const float*)d_in[0];     // (N_RAYS, 6) f32
  const int*   ri       = (const int*)  d_in[1];     // (T,) i32, sorted
  const float* t_starts = (const float*)d_in[2];     // (T,) f32
  const float* t_ends   = (const float*)d_in[3];     // (T,) f32

  const int n_rays = in_sizes[0] / 6;
  const int T      = in_sizes[1];

  // Output layout (flat, return order): xyzs | ray_valid | z_vals | dists
  float* out    = (float*)d_out;
  f4*    xyzs   = (f4*)out;
  float* valid  = out + (size_t)T * 4;
  float* z_vals = valid  + (size_t)n_rays * MAX_SAMPS;
  float* dists  = z_vals + (size_t)n_rays * MAX_SAMPS;

  int* seg_start = (int*)d_ws;
  int* seg_end   = seg_start + n_rays;

  k_init_seg  <<<(n_rays + 255) / 256, 256, 0, stream>>>(seg_start, seg_end, n_rays);

  int bnd_threads = (T + 3) / 4;
  k_boundaries<<<(bnd_threads + 255) / 256, 256, 0, stream>>>(ri, seg_start, seg_end, T);

  k_fill      <<<(n_rays + 3) / 4, 256, 0, stream>>>(t_starts, t_ends, seg_start, seg_end,
                                                     (f4*)valid, (f4*)z_vals, (f4*)dists,
                                                     n_rays);

  k_xyz       <<<(T + 255) / 256, 256, 0, stream>>>(rays, ri, t_starts, t_ends, xyzs, T);
}